// Full_46213848105991
// MI455X (gfx1250) — compile-verified
//
#include <hip/hip_runtime.h>
#include <hip/hip_bf16.h>

typedef __attribute__((ext_vector_type(2))) float v2f;
typedef __attribute__((ext_vector_type(8))) float v8f;

// One wave32 handles 16 batch elements via V_WMMA_F32_16X16X4_F32:
//   D[m,n] = sum_k first_m[k] * def_n[k];  out[m] = D[m,m] + b[a[m]]
// A-fragment (16x4 f32): lane m (0-15) = {A[m,k0],A[m,k0+1]}, lane m+16 = {A[m,k0+2],A[m,k0+3]}
// B-fragment (4x16 f32): lane n (0-15) = {B[k0,n],B[k0+1,n]} = {def_n[k0],def_n[k0+1]}, hi lanes +2
// => both fragments are a single per-lane b64 load. No LDS needed.
//
// Cache policy: W's reachable region (64*8*8*32 KiB = 128 MB) fits in the 192 MB L2
// and is re-read across batches -> default (RT) loads. def_pos (256 MB) and out (1 MB)
// are touched exactly once -> non-temporal, so the stream doesn't evict W from L2.
__global__ __launch_bounds__(256) void gather_tile_dot_wmma(
    const float* __restrict__ W,      // (64,8,32,32,16,16)
    const float* __restrict__ b,      // (64,)
    const int*   __restrict__ a,      // (BATCH,)
    const int*   __restrict__ style,  // (BATCH,)
    const int*   __restrict__ bh,     // (BATCH,2)
    const float* __restrict__ defp,   // (BATCH,16,16)
    float*       __restrict__ out)    // (BATCH,)
{
    const int lane   = threadIdx.x & 31;
    const int wave   = threadIdx.x >> 5;
    const int batch0 = (blockIdx.x * 8 + wave) * 16;
    const int m      = lane & 15;      // row (A) / col (B) index this lane serves
    const int hi     = lane >> 4;      // 0: K=k0,k0+1   1: K=k0+2,k0+3
    const int batch  = batch0 + m;

    // Per-lane gather indices (lanes 16-31 duplicate lanes 0-15's scalars).
    int ai = a[batch];
    int si = style[batch];
    int t1 = bh[2 * batch];
    int t2 = bh[2 * batch + 1];
    t1 = (t1 >= 8)  ? 7  : t1;   // reference clamps t1 against W.shape[1] == 8
    t2 = (t2 >= 32) ? 31 : t2;   // no-op clamp, kept for fidelity
    const float bias = b[ai];

    const float* wt = W + ((((size_t)ai * 8 + si) * 32 + (size_t)t1) * 32 + (size_t)t2) * 256;
    const float* dp = defp + (size_t)batch * 256;
    const int koff = hi * 2;

    v8f c = {};
#pragma unroll 8
    for (int k0 = 0; k0 < 256; k0 += 4) {
        v2f av = *(const v2f*)(wt + k0 + koff);                          // RT: keep W in L2
        v2f bv = __builtin_nontemporal_load((const v2f*)(dp + k0 + koff)); // NT: stream def_pos
        // (neg_a, A, neg_b, B, c_mod, C, reuse_a, reuse_b)
        c = __builtin_amdgcn_wmma_f32_16x16x4_f32(
                false, av, false, bv, (short)0, c, false, false);
    }

    // Diagonal extraction:
    //  lane l in [0,8):   D[l,l]       = c[l]
    //  lane l in [24,32): D[l-16,l-16] = c[l-24]
    const int j = lane & 7;
    float r = c[0];
    r = (j == 1) ? c[1] : r;
    r = (j == 2) ? c[2] : r;
    r = (j == 3) ? c[3] : r;
    r = (j == 4) ? c[4] : r;
    r = (j == 5) ? c[5] : r;
    r = (j == 6) ? c[6] : r;
    r = (j == 7) ? c[7] : r;

    if (lane < 8 || lane >= 24) {
        // For lane>=24: batch = batch0 + lane - 16, matching the diagonal row.
        __builtin_nontemporal_store(r + bias, out + batch);
    }
}

extern "C" void kernel_launch(void* const* d_in, const int* in_sizes, int n_in,
                              void* d_out, int out_size, void* d_ws, size_t ws_size,
                              hipStream_t stream) {
    const float* W     = (const float*)d_in[0];
    const float* b     = (const float*)d_in[1];
    const int*   a     = (const int*)  d_in[2];
    const int*   style = (const int*)  d_in[3];
    const int*   bh    = (const int*)  d_in[4];
    const float* defp  = (const float*)d_in[5];
    float*       out   = (float*)d_out;

    const int batch = in_sizes[2];          // 262144
    const int batches_per_block = 128;      // 8 waves x 16
    const int blocks = batch / batches_per_block;

    gather_tile_dot_wmma<<<blocks, 256, 0, stream>>>(W, b, a, style, bh, defp, out);
}